// SelfAttention_73693048864964
// MI455X (gfx1250) — compile-verified
//
#include <hip/hip_runtime.h>
#include <cmath>

typedef __attribute__((ext_vector_type(2))) float v2f;
typedef __attribute__((ext_vector_type(8))) float v8f;

constexpr int B_ = 8, C_ = 512, CQ = 64, N_ = 4096;
constexpr int FS = 72;                       // padded LDS row stride (floats): 16B-aligned rows,
                                             // half-wave fragment reads land on disjoint banks
constexpr int SMEM_FLOATS = 2 * 64 * FS      // f tiles (double buffered)
                          + 64 * FS          // g tile
                          + 64 * FS          // S / P tile
                          + 3 * 64;          // mrow, lrow, rfac
constexpr int SMEM_BYTES = SMEM_FLOATS * 4;

// ---- gfx1250 async memory->LDS copy (ASYNCcnt), inline asm: portable across toolchains.
// Low 32 bits of a flat shared pointer == LDS byte offset (LDS aperture keeps the
// offset in addr[31:0]), which is exactly what VDST of the async op expects.
__device__ __forceinline__ void async_cp16(float* lds, const float* gsrc) {
    unsigned loff = (unsigned)(unsigned long long)(uintptr_t)lds;
    unsigned long long ga = (unsigned long long)(uintptr_t)gsrc;
    asm volatile("global_load_async_to_lds_b128 %0, %1, off" :: "v"(loff), "v"(ga) : "memory");
}
__device__ __forceinline__ void async_wait0() {
    asm volatile("s_wait_asynccnt 0" ::: "memory");
}

// ---------------------------------------------------------------------------
// Kernel 1: inverse spectral norm.  sigma = ||Wv||^2/(||Wv||+eps), v=norm(W^T u).
// ---------------------------------------------------------------------------
__global__ void sigma_kernel(const float* __restrict__ W, const float* __restrict__ u,
                             int O, int Cin, float* __restrict__ inv_sigma_out) {
    __shared__ float t[512];
    __shared__ float red[256];
    const int tid = threadIdx.x;

    float local = 0.f;
    for (int cin = tid; cin < Cin; cin += 256) {
        float s = 0.f;
        for (int o = 0; o < O; ++o) s += W[(size_t)o * Cin + cin] * u[o];
        t[cin] = s;
        local += s * s;
    }
    red[tid] = local;
    __syncthreads();
    for (int s = 128; s > 0; s >>= 1) { if (tid < s) red[tid] += red[tid + s]; __syncthreads(); }
    float rn = 1.f / (sqrtf(red[0]) + 1e-12f);
    __syncthreads();
    for (int cin = tid; cin < Cin; cin += 256) t[cin] *= rn;
    __syncthreads();

    local = 0.f;
    for (int o = tid; o < O; o += 256) {
        float s = 0.f;
        for (int c = 0; c < Cin; ++c) s += W[(size_t)o * Cin + c] * t[c];
        local += s * s;
    }
    red[tid] = local;
    __syncthreads();
    for (int s = 128; s > 0; s >>= 1) { if (tid < s) red[tid] += red[tid + s]; __syncthreads(); }
    if (tid == 0) {
        float ssq = red[0];
        float q   = sqrtf(ssq);
        inv_sigma_out[0] = (q + 1e-12f) / ssq;
    }
}

// ---------------------------------------------------------------------------
// Kernel 2: projection GEMM  out[b,o,n] = inv_sigma*(W @ x_b)[o,n] + bias[o]
// ---------------------------------------------------------------------------
__global__ void proj_kernel(const float* __restrict__ W, const float* __restrict__ bias,
                            const float* __restrict__ inv_sigma,
                            const float* __restrict__ x, float* __restrict__ out, int O) {
    __shared__ float Wt[64][33];
    __shared__ float Xt[32][65];
    const int tid = threadIdx.x;
    const int tx = tid & 15, ty = tid >> 4;
    const int n0 = blockIdx.x * 64;
    const int o0 = blockIdx.y * 64;
    const int b  = blockIdx.z;

    float acc[4][4] = {};
    for (int kt = 0; kt < C_; kt += 32) {
        #pragma unroll
        for (int k = 0; k < 8; ++k) {
            int idx = tid + 256 * k;
            int r = idx >> 5, c = idx & 31;
            Wt[r][c] = W[(size_t)(o0 + r) * C_ + kt + c];
        }
        #pragma unroll
        for (int k = 0; k < 8; ++k) {
            int idx = tid + 256 * k;
            int r = idx >> 6, c = idx & 63;
            Xt[r][c] = x[((size_t)b * C_ + kt + r) * N_ + n0 + c];
        }
        __syncthreads();
        for (int k = 0; k < 32; ++k) {
            float wv[4], xv[4];
            #pragma unroll
            for (int i = 0; i < 4; ++i) wv[i] = Wt[ty + 16 * i][k];
            #pragma unroll
            for (int j = 0; j < 4; ++j) xv[j] = Xt[k][tx + 16 * j];
            #pragma unroll
            for (int i = 0; i < 4; ++i)
                #pragma unroll
                for (int j = 0; j < 4; ++j) acc[i][j] += wv[i] * xv[j];
        }
        __syncthreads();
    }
    const float is = inv_sigma[0];
    #pragma unroll
    for (int i = 0; i < 4; ++i) {
        const int o = o0 + ty + 16 * i;
        const float bb = bias[o];
        #pragma unroll
        for (int j = 0; j < 4; ++j) {
            const int n = n0 + tx + 16 * j;
            out[((size_t)b * O + o) * N_ + n] = is * acc[i][j] + bb;
        }
    }
}

// ---------------------------------------------------------------------------
// Kernel 3: fused flash attention  s=f^T g, softmax over keys i, o=h@beta,
// y = gamma*o + x.  f key-tiles double-buffered via global_load_async_to_lds.
// ---------------------------------------------------------------------------
__global__ void __launch_bounds__(256)
attn_kernel(const float* __restrict__ F, const float* __restrict__ G,
            const float* __restrict__ H, const float* __restrict__ x,
            const float* __restrict__ gamma, float* __restrict__ out) {
    extern __shared__ float smem[];
    float* fT   = smem;                 // [2][64][FS]
    float* gT   = fT + 2 * 64 * FS;     // [64][FS]
    float* ST   = gT + 64 * FS;         // [64][FS]
    float* mrow = ST + 64 * FS;         // [64]
    float* lrow = mrow + 64;            // [64]
    float* rfac = lrow + 64;            // [64]

    const int tid  = threadIdx.x;
    const int lane = tid & 31;
    const int wv   = tid >> 5;
    const int ln   = lane & 15;
    const int koff = (lane < 16) ? 0 : 2;   // K-pair held by this half-wave (A/B frags)
    const int moff = (lane < 16) ? 0 : 8;   // row offset in C/D fragment
    const int b  = blockIdx.x >> 6;
    const int j0 = (blockIdx.x & 63) * 64;

    // async-stage persistent g tile (row chunks of 16B)
    #pragma unroll
    for (int k = 0; k < 4; ++k) {
        int idx = tid + 256 * k;
        int c = idx >> 4, ch = (idx & 15) * 4;
        async_cp16(&gT[c * FS + ch], &G[((size_t)b * CQ + c) * N_ + j0 + ch]);
    }
    // async-stage f tile 0 into buffer 0
    {
        const float* src = F + (size_t)b * CQ * N_;
        #pragma unroll
        for (int k = 0; k < 4; ++k) {
            int idx = tid + 256 * k;
            int c = idx >> 4, ch = (idx & 15) * 4;
            async_cp16(&fT[c * FS + ch], src + (size_t)c * N_ + ch);
        }
    }
    if (tid < 64) { mrow[tid] = -__builtin_inff(); lrow[tid] = 0.f; }

    const v8f vzero = {};
    v8f accO[4][4];
    #pragma unroll
    for (int ct = 0; ct < 4; ++ct)
        #pragma unroll
        for (int jt = 0; jt < 4; ++jt) accO[ct][jt] = vzero;

    const float* hbase = H + ((size_t)b * C_ + wv * 64) * N_;

    for (int t = 0; t < N_ / 64; ++t) {
        const int i0  = t * 64;
        float* fbuf   = fT + (t & 1) * (64 * FS);
        async_wait0();          // current tile (and initial g) landed in LDS
        __syncthreads();        // all waves' async chunks visible; prev S/P consumers done

        // prefetch next f tile into the other buffer (overlaps with compute below)
        if (t + 1 < N_ / 64) {
            float* dst = fT + ((t + 1) & 1) * (64 * FS);
            const float* src = F + (size_t)b * CQ * N_ + (t + 1) * 64;
            #pragma unroll
            for (int k = 0; k < 4; ++k) {
                int idx = tid + 256 * k;
                int c = idx >> 4, ch = (idx & 15) * 4;
                async_cp16(&dst[c * FS + ch], src + (size_t)c * N_ + ch);
            }
        }

        // S[i][j] = sum_c f[c][i]*g[c][j] : 16 subtiles, 2 per wave
        #pragma unroll
        for (int sIdx = 0; sIdx < 2; ++sIdx) {
            const int s  = wv * 2 + sIdx;
            const int it = s >> 2, jt = s & 3;
            v8f acc = vzero;
            for (int c4 = 0; c4 < 64; c4 += 4) {
                v2f a, bb;
                a.x  = fbuf[(c4 + koff    ) * FS + it * 16 + ln]; // A = f^T (M=i, K=c)
                a.y  = fbuf[(c4 + koff + 1) * FS + it * 16 + ln];
                bb.x = gT  [(c4 + koff    ) * FS + jt * 16 + ln]; // B = g   (K=c, N=j)
                bb.y = gT  [(c4 + koff + 1) * FS + jt * 16 + ln];
                acc = __builtin_amdgcn_wmma_f32_16x16x4_f32(false, a, false, bb,
                                                            (short)0, acc, false, false);
            }
            #pragma unroll
            for (int r = 0; r < 8; ++r)
                ST[(it * 16 + r + moff) * FS + jt * 16 + ln] = acc[r];
        }
        __syncthreads();

        // online softmax over keys i (per column j)
        if (tid < 64) {
            const int j = tid;
            float cm = -__builtin_inff();
            for (int i = 0; i < 64; ++i) cm = fmaxf(cm, ST[i * FS + j]);
            const float mo = mrow[j];
            const float mn = fmaxf(mo, cm);
            rfac[j] = __expf(mo - mn);
            mrow[j] = mn;
        }
        __syncthreads();
        #pragma unroll
        for (int k = 0; k < 16; ++k) {
            int idx = tid + 256 * k;
            int i = idx >> 6, j = idx & 63;
            ST[i * FS + j] = __expf(ST[i * FS + j] - mrow[j]);    // P
        }
        __syncthreads();
        if (tid < 64) {
            const int j = tid;
            float cs = 0.f;
            for (int i = 0; i < 64; ++i) cs += ST[i * FS + j];
            lrow[j] = lrow[j] * rfac[j] + cs;
        }

        // rescale running o, then o += h_tile @ P  (rank-64 update, K steps of 4)
        #pragma unroll
        for (int jt = 0; jt < 4; ++jt) {
            const float fac = rfac[jt * 16 + ln];
            #pragma unroll
            for (int ct = 0; ct < 4; ++ct)
                #pragma unroll
                for (int r = 0; r < 8; ++r) accO[ct][jt][r] *= fac;
        }
        for (int kk = 0; kk < 16; ++kk) {
            const int ib = kk * 4;
            v2f hA[4], pB[4];
            #pragma unroll
            for (int ct = 0; ct < 4; ++ct) {        // A = h (M=c, K=i)
                const float* hp = hbase + (size_t)(ct * 16 + ln) * N_ + i0 + ib + koff;
                hA[ct].x = hp[0];
                hA[ct].y = hp[1];
            }
            #pragma unroll
            for (int jt = 0; jt < 4; ++jt) {        // B = P (K=i, N=j)
                pB[jt].x = ST[(ib + koff    ) * FS + jt * 16 + ln];
                pB[jt].y = ST[(ib + koff + 1) * FS + jt * 16 + ln];
            }
            #pragma unroll
            for (int ct = 0; ct < 4; ++ct)
                #pragma unroll
                for (int jt = 0; jt < 4; ++jt)
                    accO[ct][jt] = __builtin_amdgcn_wmma_f32_16x16x4_f32(
                        false, hA[ct], false, pB[jt], (short)0, accO[ct][jt], false, false);
        }
    }
    __syncthreads();

    // epilogue: y = gamma * o / l + x
    const float gm = gamma[0];
    float linv[4];
    #pragma unroll
    for (int jt = 0; jt < 4; ++jt) linv[jt] = 1.f / lrow[jt * 16 + ln];
    #pragma unroll
    for (int ct = 0; ct < 4; ++ct)
        #pragma unroll
        for (int jt = 0; jt < 4; ++jt)
            #pragma unroll
            for (int r = 0; r < 8; ++r) {
                const int c = wv * 64 + ct * 16 + r + moff;
                const int j = j0 + jt * 16 + ln;
                const size_t idx = ((size_t)b * C_ + c) * N_ + j;
                out[idx] = gm * (accO[ct][jt][r] * linv[jt]) + x[idx];
            }
}

// ---------------------------------------------------------------------------
extern "C" void kernel_launch(void* const* d_in, const int* in_sizes, int n_in,
                              void* d_out, int out_size, void* d_ws, size_t ws_size,
                              hipStream_t stream) {
    (void)in_sizes; (void)n_in; (void)out_size; (void)ws_size;
    const float* x     = (const float*)d_in[0];
    const float* Wf    = (const float*)d_in[1];
    const float* bf    = (const float*)d_in[2];
    const float* Wg    = (const float*)d_in[3];
    const float* bg    = (const float*)d_in[4];
    const float* Wh    = (const float*)d_in[5];
    const float* bh    = (const float*)d_in[6];
    const float* gamma = (const float*)d_in[7];
    const float* uf    = (const float*)d_in[8];
    const float* ug    = (const float*)d_in[9];
    const float* uh    = (const float*)d_in[10];
    float* out = (float*)d_out;

    // workspace: [inv_sigma x3 (pad 16)] [f: B*CQ*N] [g: B*CQ*N] [h: B*C*N]
    float* wsf = (float*)d_ws;
    float* inv = wsf;
    float* Fb  = wsf + 16;
    float* Gb  = Fb + (size_t)B_ * CQ * N_;
    float* Hb  = Gb + (size_t)B_ * CQ * N_;

    static int smem_attr_set = 0;  // deterministic: same call every time, idempotent
    if (!smem_attr_set) {
        hipFuncSetAttribute((const void*)attn_kernel,
                            hipFuncAttributeMaxDynamicSharedMemorySize, SMEM_BYTES);
        smem_attr_set = 1;
    }

    sigma_kernel<<<1, 256, 0, stream>>>(Wf, uf, CQ, C_, inv + 0);
    sigma_kernel<<<1, 256, 0, stream>>>(Wg, ug, CQ, C_, inv + 1);
    sigma_kernel<<<1, 256, 0, stream>>>(Wh, uh, C_, C_, inv + 2);

    proj_kernel<<<dim3(N_ / 64, CQ / 64, B_), 256, 0, stream>>>(Wf, bf, inv + 0, x, Fb, CQ);
    proj_kernel<<<dim3(N_ / 64, CQ / 64, B_), 256, 0, stream>>>(Wg, bg, inv + 1, x, Gb, CQ);
    proj_kernel<<<dim3(N_ / 64, C_ / 64, B_), 256, 0, stream>>>(Wh, bh, inv + 2, x, Hb, C_);

    attn_kernel<<<dim3(B_ * (N_ / 64)), 256, SMEM_BYTES, stream>>>(Fb, Gb, Hb, x, gamma, out);
}